// TorchsparseMaxPool_63496796504722
// MI455X (gfx1250) — compile-verified
//
#include <hip/hip_runtime.h>

#define NEG_FILL (-1000.0f)

// ---------------------------------------------------------------------------
// Hash table: open addressing, linear probe. Keys are unique non-negative
// int32 (max ~4.4e8), so -1 is a safe empty sentinel. 2^20 entries for 400k
// keys -> load factor ~0.38, expected probes ~1.3.
// ---------------------------------------------------------------------------
static constexpr unsigned TAB_DEFAULT = 1u << 20;

__device__ __forceinline__ unsigned hash_key(int key, unsigned mask) {
    return ((unsigned)key * 2654435761u) & mask;  // Knuth multiplicative
}

__global__ void init_table_kernel(int* __restrict__ keys, unsigned tab) {
    unsigned i = blockIdx.x * blockDim.x + threadIdx.x;
    if (i < tab) keys[i] = -1;
}

__global__ void build_table_kernel(const int4* __restrict__ coords,
                                   int* __restrict__ keys,
                                   int* __restrict__ vals,
                                   int N, unsigned mask) {
    int i = blockIdx.x * blockDim.x + threadIdx.x;
    if (i >= N) return;
    int4 c = coords[i];  // (x, y, z, b)
    int key = ((c.w * 512 + c.x) * 512 + c.y) * 512 + c.z;
    unsigned h = hash_key(key, mask);
    while (true) {
        int prev = atomicCAS(&keys[h], -1, key);
        if (prev == -1) { vals[h] = i; break; }
        if (prev == key) break;  // defensive: keys are unique per call
        h = (h + 1) & mask;
    }
}

// One wave32 per output row: lanes 0..7 resolve the 8 children via the hash
// table and prefetch their feature rows; all 32 lanes gather float2 (2
// channels each, 256B coalesced per feature row) and max-reduce.
__global__ void pool_kernel(const int4* __restrict__ out_coords,
                            const float* __restrict__ feats,
                            const int* __restrict__ keys,
                            const int* __restrict__ vals,
                            float* __restrict__ out,
                            int M, int C, unsigned mask) {
    int gid  = blockIdx.x * blockDim.x + threadIdx.x;
    int wave = gid >> 5;
    int lane = gid & 31;
    if (wave >= M) return;

    int4 oc = out_coords[wave];
    int row = -1;
    if (lane < 8) {
        int x = oc.x * 2 + ((lane >> 2) & 1);
        int y = oc.y * 2 + ((lane >> 1) & 1);
        int z = oc.z * 2 + (lane & 1);
        int key = ((oc.w * 512 + x) * 512 + y) * 512 + z;
        unsigned h = hash_key(key, mask);
        while (true) {
            int k = keys[h];
            if (k == key) { row = vals[h]; break; }
            if (k == -1) break;
            h = (h + 1) & mask;
        }
        if (row >= 0) {
            // gfx1250: global_prefetch_b8 — start pulling the feature row
            // (256B = 2 cachelines) while the wave does the shuffles below.
            const float* p = feats + (size_t)row * C;
            __builtin_prefetch(p, 0, 3);
            __builtin_prefetch(p + 32, 0, 3);
        }
    }

    // Broadcast the 8 resolved rows across the wave (wave32 ds_bpermute).
    int rows[8];
#pragma unroll
    for (int j = 0; j < 8; ++j) rows[j] = __shfl(row, j, 32);

    // Channel-wise max. C == 64 in practice: one float2 per lane.
    for (int c = lane * 2; c < C; c += 64) {
        float2 m = make_float2(NEG_FILL, NEG_FILL);
#pragma unroll
        for (int j = 0; j < 8; ++j) {
            int r = rows[j];
            if (r >= 0) {
                const float2 f =
                    *(const float2*)(feats + (size_t)r * C + c);
                m.x = fmaxf(m.x, f.x);
                m.y = fmaxf(m.y, f.y);
            }
        }
        *(float2*)(out + (size_t)wave * C + c) = m;
    }
}

extern "C" void kernel_launch(void* const* d_in, const int* in_sizes, int n_in,
                              void* d_out, int out_size, void* d_ws, size_t ws_size,
                              hipStream_t stream) {
    const int4*  coords = (const int4*)d_in[0];   // (N,4) int32
    const float* feats  = (const float*)d_in[1];  // (N,C) f32
    const int4*  outc   = (const int4*)d_in[2];   // (M,4) int32
    float*       out    = (float*)d_out;          // (M,C) f32

    const int N = in_sizes[0] / 4;
    const int M = in_sizes[2] / 4;
    const int C = (N > 0) ? (in_sizes[1] / N) : 64;  // 64 for this problem

    // Fit the table in the provided workspace (keys + vals arrays).
    unsigned tab = TAB_DEFAULT;
    while ((size_t)tab * 2u * sizeof(int) > ws_size && tab > 2u) tab >>= 1;
    const unsigned mask = tab - 1;
    int* keys = (int*)d_ws;
    int* vals = keys + tab;

    init_table_kernel<<<(tab + 255) / 256, 256, 0, stream>>>(keys, tab);
    build_table_kernel<<<(N + 255) / 256, 256, 0, stream>>>(coords, keys, vals, N, mask);

    const long threads = (long)M * 32;
    pool_kernel<<<(threads + 255) / 256, 256, 0, stream>>>(outc, feats, keys, vals,
                                                           out, M, C, mask);
}